// TransportOperatorHeader_83975200571941
// MI455X (gfx1250) — compile-verified
//
#include <hip/hip_runtime.h>
#include <math.h>

// ---------------------------------------------------------------------------
// Transport-operator FISTA inference on MI455X (gfx1250), f32 WMMA path.
// B=2048, N=32, M=100.  All matrices are tiny; problem is L2-resident.
// Core op: chains of 32x32 f32 matmuls -> V_WMMA_F32_16X16X4_F32, one wave
// per batch element, matrices staged in LDS.
// ---------------------------------------------------------------------------

typedef __attribute__((ext_vector_type(2))) float v2f;
typedef __attribute__((ext_vector_type(8))) float v8f;

#define B_SZ   2048
#define N_DIM  32
#define M_DIM  100
#define NN     1024          // N*N
#define LR     0.01f
#define ZETA   0.1f
#define N_ITER 10
#define K_TAY  10            // Taylor order; ||A|| ~ 0.05 -> error << f32 eps

// ---- WMMA f32 16x16x4 fragment loaders (wave32) ---------------------------
//   A (16x4): VGPR0 lanes0-15 (M=lane,K=k0), lanes16-31 (M=lane-16,K=k0+2);
//             VGPR1 same with K+1.
//   B (4x16): VGPR0 lanes0-15 (K=k0,N=lane), lanes16-31 (K=k0+2,N=lane-16);
//             VGPR1 same with K+1.
//   C (16x16): VGPR v: lanes0-15 (M=v,N=lane), lanes16-31 (M=v+8,N=lane-16).

static __device__ __forceinline__ v2f frag_a(const float* buf, int lane,
                                             int ti, int kb, bool trans,
                                             float scale) {
  int r  = ti * 16 + (lane & 15);
  int k0 = kb * 4 + ((lane >> 4) << 1);
  float a0, a1;
  if (!trans) { a0 = buf[r * 32 + k0];  a1 = buf[r * 32 + k0 + 1]; }
  else        { a0 = buf[k0 * 32 + r];  a1 = buf[(k0 + 1) * 32 + r]; } // Aᵀ
  v2f o; o.x = a0 * scale; o.y = a1 * scale; return o;
}

static __device__ __forceinline__ v2f frag_b(const float* buf, int lane,
                                             int tj, int kb, bool trans) {
  int c  = tj * 16 + (lane & 15);
  int k0 = kb * 4 + ((lane >> 4) << 1);
  float b0, b1;
  if (!trans) { b0 = buf[k0 * 32 + c];  b1 = buf[(k0 + 1) * 32 + c]; }
  else        { b0 = buf[c * 32 + k0];  b1 = buf[c * 32 + k0 + 1]; }  // Bᵀ
  v2f o; o.x = b0; o.y = b1; return o;
}

// dst = scale*(opA x opB) [+ dst], all 32x32 f32 in LDS, one wave.
static __device__ void wave_mm(float* dst, const float* Am, bool tA,
                               const float* Bm, bool tB, float scale,
                               bool acc, int lane) {
  for (int ti = 0; ti < 2; ++ti) {
    for (int tj = 0; tj < 2; ++tj) {
      v8f c = {};
      if (acc) {
        for (int v = 0; v < 8; ++v) {
          int r  = ti * 16 + v + ((lane >> 4) << 3);
          int cc = tj * 16 + (lane & 15);
          c[v] = dst[r * 32 + cc];
        }
      }
      for (int kb = 0; kb < 8; ++kb) {
        v2f a = frag_a(Am, lane, ti, kb, tA, scale);
        v2f b = frag_b(Bm, lane, tj, kb, tB);
        c = __builtin_amdgcn_wmma_f32_16x16x4_f32(false, a, false, b,
                                                  (short)0, c, false, false);
      }
      for (int v = 0; v < 8; ++v) {
        int r  = ti * 16 + v + ((lane >> 4) << 3);
        int cc = tj * 16 + (lane & 15);
        dst[r * 32 + cc] = c[v];
      }
    }
  }
  __syncthreads(); // single-wave WG: ~free; orders LDS RAW
}

// ---- Kernel 1: A[b,:] = Y[b,:] x PSI  ([B,M]x[M,NN]) ----------------------
__global__ void coeff_psi_gemm(const float* __restrict__ Y,
                               const float* __restrict__ psi,
                               float* __restrict__ Aout) {
  int bi = blockIdx.x;           // 0..127  -> rows bi*16..bi*16+15
  int j  = blockIdx.y;           // 0..63   -> cols j*16..j*16+15
  int lane = threadIdx.x;
  int row = bi * 16 + (lane & 15);
  int col = j * 16 + (lane & 15);
  v8f c = {};
  for (int kb = 0; kb < 25; ++kb) {       // M=100 = 25*4 exactly
    int k0 = kb * 4 + ((lane >> 4) << 1);
    v2f a; a.x = Y[row * M_DIM + k0]; a.y = Y[row * M_DIM + k0 + 1];
    v2f b; b.x = psi[k0 * NN + col];  b.y = psi[(k0 + 1) * NN + col];
    c = __builtin_amdgcn_wmma_f32_16x16x4_f32(false, a, false, b,
                                              (short)0, c, false, false);
  }
  for (int v = 0; v < 8; ++v) {
    int r = bi * 16 + v + ((lane >> 4) << 3);
    Aout[(size_t)r * NN + j * 16 + (lane & 15)] = c[v];
  }
}

// ---- Kernel 2: per-batch expm (Taylor) fwd + adjoint Fréchet bwd ----------
// One wave (32 threads) per batch element; 7x 32x32 f32 matrices in LDS.
// Buffer roles (by index into buf[7]):
//   0: A   (always)
//   fwd : 1,2 = P ping-pong, 3 = T
//   bwd : 4 = G, 5 = Gc, 1,2 = pp ping-pong, 3,6 = q ping-pong
__global__ void expm_grad_kernel(const float* __restrict__ Ag,
                                 const float* __restrict__ z0g,
                                 const float* __restrict__ z1g,
                                 float* __restrict__ Gcg,
                                 float* __restrict__ z1hat,
                                 int forward_only) {
  __shared__ float buf[7][NN];
  __shared__ float zz0[N_DIM], rr[N_DIM];
  int b = blockIdx.x;
  int lane = threadIdx.x;

  float* sA = buf[0];
  for (int i = lane; i < NN; i += 32) sA[i] = Ag[(size_t)b * NN + i];
  zz0[lane] = z0g[b * N_DIM + lane];
  __syncthreads();

  // ---- forward: T = I + A + sum_{k>=2} A^k/k!  (P_k = A*P_{k-1}/k) ----
  float* T = buf[3];
  for (int i = lane; i < NN; i += 32)
    T[i] = sA[i] + (((i >> 5) == (i & 31)) ? 1.0f : 0.0f);
  __syncthreads();

  {
    int pold = 0;                        // buf index (0 == A = P_1)
    int sel = 0;
    for (int k = 2; k <= K_TAY; ++k) {
      float* pnew = buf[1 + sel];        // runtime-indexed LDS pointer
      wave_mm(pnew, sA, false, buf[pold], false, 1.0f / (float)k, false, lane);
      for (int i = lane; i < NN; i += 32) T[i] += pnew[i];
      __syncthreads();
      pold = 1 + sel;
      sel ^= 1;
    }
  }

  // ---- r = T z0 - z1 ; or emit z1_hat = T z0 ----
  float dot = 0.0f;
  for (int q = 0; q < N_DIM; ++q) dot += T[lane * N_DIM + q] * zz0[q];
  if (forward_only) {
    z1hat[b * N_DIM + lane] = dot;       // LATENT_SCALE == 1
    return;
  }
  rr[lane] = dot - z1g[b * N_DIM + lane];
  __syncthreads();

  // ---- backward: G = r z0ᵀ ; dL/dA = L_expm(Aᵀ, G) via series ----
  // q_1 = G; q_k = (G pp_{k-1} + Aᵀ q_{k-1})/k ; pp_k = Aᵀ pp_{k-1}/k (pp_1=Aᵀ)
  float* G  = buf[4];
  float* Gc = buf[5];
  for (int i = lane; i < NN; i += 32) {
    float g = rr[i >> 5] * zz0[i & 31];
    G[i]  = g;
    Gc[i] = g;                           // k = 1 term
  }
  __syncthreads();

  {
    int qold  = 4;                       // G
    int ppold = 0;                       // A, read transposed while k==2
    int qs = 0, ps = 0;
    for (int k = 2; k <= K_TAY; ++k) {
      float inv = 1.0f / (float)k;
      bool first = (k == 2);
      int qn_idx = 3 + 3 * qs;           // 3 or 6
      float* qn = buf[qn_idx];
      wave_mm(qn, G, false, buf[ppold], first /*ppold==Aᵀ*/, inv, false, lane);
      wave_mm(qn, sA, true, buf[qold], false, inv, true, lane);
      for (int i = lane; i < NN; i += 32) Gc[i] += qn[i];
      __syncthreads();
      if (k < K_TAY) {
        int pn_idx = 1 + ps;             // 1 or 2
        wave_mm(buf[pn_idx], sA, true, buf[ppold], first, inv, false, lane);
        ppold = pn_idx;
        ps ^= 1;
      }
      qold = qn_idx;
      qs ^= 1;
    }
  }

  for (int i = lane; i < NN; i += 32) Gcg[(size_t)b * NN + i] = Gc[i];
}

// ---- Kernel 3: g = Gc x PSIᵀ ([B,NN]x[NN,M]) + fused FISTA update ---------
__global__ void grad_update_kernel(const float* __restrict__ Gc,
                                   const float* __restrict__ psi,
                                   float* __restrict__ c,
                                   float* __restrict__ y,
                                   float mu) {
  int bi = blockIdx.x;           // 0..127
  int mi = blockIdx.y;           // 0..6  (ceil(100/16))
  int lane = threadIdx.x;
  int row = bi * 16 + (lane & 15);
  int col = mi * 16 + (lane & 15);
  bool colok = (col < M_DIM);
  v8f acc = {};
  for (int kb = 0; kb < 256; ++kb) {      // K = 1024
    int k0 = kb * 4 + ((lane >> 4) << 1);
    v2f a; a.x = Gc[(size_t)row * NN + k0];
           a.y = Gc[(size_t)row * NN + k0 + 1];
    v2f b;                                 // PSIᵀ[k, m] = psi[m*NN + k]
    b.x = colok ? psi[(size_t)col * NN + k0]     : 0.0f;
    b.y = colok ? psi[(size_t)col * NN + k0 + 1] : 0.0f;
    acc = __builtin_amdgcn_wmma_f32_16x16x4_f32(false, a, false, b,
                                                (short)0, acc, false, false);
  }
  if (!colok) return;
  const float th = LR * ZETA;
  for (int v = 0; v < 8; ++v) {
    int r = bi * 16 + v + ((lane >> 4) << 3);
    size_t idx = (size_t)r * M_DIM + col;
    float g  = acc[v];
    float yo = y[idx];
    float co = c[idx];
    float u  = yo - LR * g;
    float cn = copysignf(fmaxf(fabsf(u) - th, 0.0f), u);
    c[idx] = cn;
    y[idx] = cn + mu * (cn - co);
  }
}

// ---- init: FISTA state from c_init; pass-through outputs z0,z1,z1 ---------
__global__ void init_kernel(const float* __restrict__ z0,
                            const float* __restrict__ z1,
                            const float* __restrict__ c_init,
                            float* __restrict__ c, float* __restrict__ y,
                            float* __restrict__ out) {
  int i = blockIdx.x * blockDim.x + threadIdx.x;
  if (i < B_SZ * M_DIM) { c[i] = c_init[i]; y[i] = c_init[i]; }
  if (i < B_SZ * N_DIM) {
    out[i] = z0[i];
    out[B_SZ * N_DIM + i] = z1[i];
    out[3 * B_SZ * N_DIM + i] = z1[i];
  }
}

extern "C" void kernel_launch(void* const* d_in, const int* in_sizes, int n_in,
                              void* d_out, int out_size, void* d_ws,
                              size_t ws_size, hipStream_t stream) {
  const float* z0     = (const float*)d_in[0];   // feature_0 [B,N]
  const float* z1     = (const float*)d_in[1];   // feature_1 [B,N]
  const float* psi    = (const float*)d_in[2];   // [M,N,N]
  const float* c_init = (const float*)d_in[3];   // [B,M]
  float* out = (float*)d_out;                    // z0|z1|z1_hat|z1

  // workspace carve-up
  char* ws = (char*)d_ws;
  float* A  = (float*)(ws);                              // [B,NN]  8 MB
  float* Gc = (float*)(ws + (size_t)B_SZ * NN * 4);      // [B,NN]  8 MB
  float* c  = (float*)(ws + (size_t)2 * B_SZ * NN * 4);  // [B,M]
  float* y  = (float*)(ws + (size_t)2 * B_SZ * NN * 4 + (size_t)B_SZ * M_DIM * 4);

  // FISTA momentum coefficients are data-independent: precompute on host.
  float mus[N_ITER];
  {
    float t = 1.0f;
    for (int i = 0; i < N_ITER; ++i) {
      float tn = 0.5f * (1.0f + sqrtf(1.0f + 4.0f * t * t));
      mus[i] = (t - 1.0f) / tn;
      t = tn;
    }
  }

  init_kernel<<<(B_SZ * M_DIM + 255) / 256, 256, 0, stream>>>(z0, z1, c_init,
                                                              c, y, out);

  dim3 g1(B_SZ / 16, NN / 16);           // 128 x 64
  dim3 g3(B_SZ / 16, (M_DIM + 15) / 16); // 128 x 7
  for (int it = 0; it < N_ITER; ++it) {
    coeff_psi_gemm<<<g1, 32, 0, stream>>>(y, psi, A);
    expm_grad_kernel<<<B_SZ, 32, 0, stream>>>(A, z0, z1, Gc, nullptr, 0);
    grad_update_kernel<<<g3, 32, 0, stream>>>(Gc, psi, c, y, mus[it]);
  }

  // final forward with converged c: z1_hat = expm(sum c_m psi_m) z0
  coeff_psi_gemm<<<g1, 32, 0, stream>>>(c, psi, A);
  expm_grad_kernel<<<B_SZ, 32, 0, stream>>>(A, z0, z1, nullptr,
                                            out + 2 * B_SZ * N_DIM, 1);
}